// Decoder_13091060318474
// MI455X (gfx1250) — compile-verified
//
#include <hip/hip_runtime.h>
#include <hip/hip_bf16.h>
#include <math.h>

// Problem dims
#define Bn 128
#define Ln 1024
#define Hn 256
#define En 128
#define Vn 50000
#define Xn 100

#define WPB 8   // waves per GEMM block (wave32 -> 256 threads)

typedef __attribute__((ext_vector_type(2))) float v2f;
typedef __attribute__((ext_vector_type(8))) float v8f;

__device__ __forceinline__ float sigmoidf_(float x) {
  return 1.0f / (1.0f + __expf(-x));
}

// ---------------------------------------------------------------------------
// Generic fp32 WMMA GEMM:  C[M,N] = (acc? C : 0) + A[M,K] @ W[N,K]^T + bias[N]
// Block = 8 waves. Block stages the 16xK A strip in LDS (ds_load_b64 feeds the
// WMMA A operand, no LOADcnt), each wave owns one 16x16 tile of C and streams
// its 16xK strip of W from global, software-pipelined one K-step ahead so the
// s_wait_loadcnt before each V_WMMA_F32_16X16X4_F32 covers loads issued a full
// WMMA earlier.
// A-matrix layout (16x4 f32): lane = {m = lane&15, kpair = 2*(lane>>4)}, 2 VGPRs.
// B-matrix layout (4x16 f32): lane = {n = lane&15, kpair = 2*(lane>>4)}, 2 VGPRs.
// C/D (16x16 f32): VGPR v holds row m = v + 8*(lane>>4), col n = lane&15.
// ---------------------------------------------------------------------------
__global__ __launch_bounds__(WPB * 32) void gemm_wmma_f32(
    const float* __restrict__ A, const float* __restrict__ W,
    const float* __restrict__ bias, float* __restrict__ C,
    int M, int N, int K, int accumulate)
{
  extern __shared__ float sA[];                 // [16][K]
  const int t    = threadIdx.x;
  const int lane = t & 31;
  const int wave = t >> 5;
  const int m0   = blockIdx.y << 4;

  // cooperative stage of A strip [m0..m0+15, 0..K) into LDS
  for (int r = 0; r < 16; ++r) {
    const float* __restrict__ src = A + (size_t)(m0 + r) * K;
    float* __restrict__ dst = sA + r * K;
    for (int c = t; c < K; c += WPB * 32) dst[c] = src[c];
  }
  __syncthreads();

  const int ntile = blockIdx.x * WPB + wave;    // uniform per wave
  if (ntile * 16 >= N) return;                  // whole wave exits together

  const int lm   = lane & 15;
  const int half = lane >> 4;                   // 0: K=0,1   1: K=2,3
  const int n0   = ntile << 4;

  const float* __restrict__ Arow = sA + lm * K + 2 * half;          // LDS
  const float* __restrict__ Wrow = W + (size_t)(n0 + lm) * K + 2 * half;

  v8f acc = {};
  v2f a0 = *reinterpret_cast<const v2f*>(Arow);   // ds_load_b64
  v2f b0 = *reinterpret_cast<const v2f*>(Wrow);   // global_load_b64
  for (int k = 0; k < K - 4; k += 4) {
    const v2f a1 = *reinterpret_cast<const v2f*>(Arow + k + 4);
    const v2f b1 = *reinterpret_cast<const v2f*>(Wrow + k + 4);
    acc = __builtin_amdgcn_wmma_f32_16x16x4_f32(
        false, a0, false, b0, (short)0, acc, false, false);
    a0 = a1;
    b0 = b1;
  }
  acc = __builtin_amdgcn_wmma_f32_16x16x4_f32(
      false, a0, false, b0, (short)0, acc, false, false);

  const int n = n0 + lm;
  const float bv = bias ? bias[n] : 0.0f;
#pragma unroll
  for (int v = 0; v < 8; ++v) {
    const int m = m0 + v + 8 * half;
    float val = acc[v] + bv;
    const size_t idx = (size_t)m * N + n;
    if (accumulate) val += C[idx];
    C[idx] = val;
  }
}

// ---------------------------------------------------------------------------
// xcat = [c_t_1 (512) | embed_W[y_t] (128)]   -> [B, 640]
// ---------------------------------------------------------------------------
__global__ __launch_bounds__(256) void build_xcat_kernel(
    const int* __restrict__ y_t, const float* __restrict__ c_t_1,
    const float* __restrict__ embed_W, float* __restrict__ xcat)
{
  const int b = blockIdx.x;
  const int row = y_t[b];
  for (int j = threadIdx.x; j < 640; j += 256) {
    float v = (j < 512) ? c_t_1[b * 512 + j]
                        : embed_W[(size_t)row * En + (j - 512)];
    xcat[b * 640 + j] = v;
  }
}

// ---------------------------------------------------------------------------
// LSTM cell (torch gate order i,f,g,o) + s_t_hat = [h | c]
// ---------------------------------------------------------------------------
__global__ __launch_bounds__(256) void lstm_kernel(
    const float* __restrict__ gates, const float* __restrict__ c0,
    float* __restrict__ h_o, float* __restrict__ c_o, float* __restrict__ sth)
{
  const int idx = blockIdx.x * 256 + threadIdx.x;    // B*H
  const int b = idx >> 8, j = idx & 255;
  const float* g = gates + (size_t)b * 1024;
  const float ig = sigmoidf_(g[j]);
  const float fg = sigmoidf_(g[256 + j]);
  const float gg = tanhf(g[512 + j]);
  const float og = sigmoidf_(g[768 + j]);
  const float c = fg * c0[idx] + ig * gg;
  const float h = og * tanhf(c);
  h_o[idx] = h;
  c_o[idx] = c;
  sth[b * 512 + j] = h;
  sth[b * 512 + 256 + j] = c;
}

// ---------------------------------------------------------------------------
// Coverage attention, fused per batch row: energy -> masked softmax -> a,
// coverage_next, c_t.  Block = 256 threads = 8 wave32.
// ---------------------------------------------------------------------------
__global__ __launch_bounds__(256) void attn_kernel(
    const float* __restrict__ enc_feats,   // [B*L, 512]
    const float* __restrict__ enc_outs,    // [B, L, 512]
    const float* __restrict__ mask,        // [B, L]
    const float* __restrict__ coverage,    // [B, L]
    const float* __restrict__ dec_feats,   // [B, 512]
    const float* __restrict__ Wc,          // [512]
    const float* __restrict__ vW,          // [512]
    float* __restrict__ a_out,             // [B, L]
    float* __restrict__ ct_out,            // [B, 512]
    float* __restrict__ cov_out)           // [B, L]
{
  __shared__ float sdec[512], swc[512], sv[512];
  __shared__ float se[Ln];
  __shared__ float sred[256];
  const int b = blockIdx.x, t = threadIdx.x;

  for (int j = t; j < 512; j += 256) {
    sdec[j] = dec_feats[b * 512 + j];
    swc[j]  = Wc[j];
    sv[j]   = vW[j];
  }
  __syncthreads();

  const int wave = t >> 5, lane = t & 31;
  for (int l = wave; l < Ln; l += 8) {
    const float cov = coverage[b * Ln + l];
    const float* ef = enc_feats + ((size_t)b * Ln + l) * 512;
    float p = 0.0f;
    for (int d = lane; d < 512; d += 32)
      p += sv[d] * tanhf(ef[d] + sdec[d] + cov * swc[d]);
    for (int off = 16; off > 0; off >>= 1) p += __shfl_xor(p, off, 32);
    if (lane == 0) se[l] = p;
  }
  __syncthreads();

  // masked softmax (unmasked Z cancels in the renormalization)
  float lmax = -3.402823466e38f;
  for (int l = t; l < Ln; l += 256) lmax = fmaxf(lmax, se[l]);
  sred[t] = lmax; __syncthreads();
  for (int s = 128; s > 0; s >>= 1) {
    if (t < s) sred[t] = fmaxf(sred[t], sred[t + s]);
    __syncthreads();
  }
  const float mx = sred[0]; __syncthreads();

  float lsum = 0.0f;
  for (int l = t; l < Ln; l += 256) {
    const float e = __expf(se[l] - mx) * mask[b * Ln + l];
    se[l] = e;
    lsum += e;
  }
  sred[t] = lsum; __syncthreads();
  for (int s = 128; s > 0; s >>= 1) {
    if (t < s) sred[t] += sred[t + s];
    __syncthreads();
  }
  const float inv = 1.0f / sred[0]; __syncthreads();

  for (int l = t; l < Ln; l += 256) {
    const float a = se[l] * inv;
    se[l] = a;
    a_out[b * Ln + l]   = a;
    cov_out[b * Ln + l] = coverage[b * Ln + l] + a;
  }
  __syncthreads();

  for (int d = t; d < 512; d += 256) {
    float acc = 0.0f;
    const float* eo = enc_outs + (size_t)b * Ln * 512 + d;
    for (int l = 0; l < Ln; ++l) acc += se[l] * eo[(size_t)l * 512];
    ct_out[b * 512 + d] = acc;
  }
}

// ---------------------------------------------------------------------------
// p_gen = sigmoid([c_t | s_t_hat | x] . pgen_W + pgen_b)
// ---------------------------------------------------------------------------
__global__ __launch_bounds__(256) void pgen_kernel(
    const float* __restrict__ ct, const float* __restrict__ sth,
    const float* __restrict__ x, const float* __restrict__ pW,
    const float* __restrict__ pb, float* __restrict__ pg_out)
{
  __shared__ float sred[256];
  const int b = blockIdx.x, t = threadIdx.x;
  float s = 0.0f;
  for (int j = t; j < 1152; j += 256) {
    const float v = (j < 512)  ? ct[b * 512 + j]
                  : (j < 1024) ? sth[b * 512 + (j - 512)]
                               : x[b * 128 + (j - 1024)];
    s += pW[j] * v;
  }
  sred[t] = s; __syncthreads();
  for (int sh = 128; sh > 0; sh >>= 1) {
    if (t < sh) sred[t] += sred[t + sh];
    __syncthreads();
  }
  if (t == 0) pg_out[b] = sigmoidf_(sred[0] + pb[0]);
}

// pcat = [h (256) | c_t (512)]  -> [B, 768]
__global__ __launch_bounds__(256) void build_pcat_kernel(
    const float* __restrict__ h, const float* __restrict__ ct,
    float* __restrict__ pcat)
{
  const int b = blockIdx.x;
  for (int j = threadIdx.x; j < 768; j += 256)
    pcat[b * 768 + j] = (j < 256) ? h[b * 256 + j] : ct[b * 512 + (j - 256)];
}

// ---------------------------------------------------------------------------
// Fused: softmax over V, scale by p_gen, write extended dist, pointer scatter
// ---------------------------------------------------------------------------
__global__ __launch_bounds__(256) void final_kernel(
    const float* __restrict__ logits,  // [B, V]
    const float* __restrict__ pgen,    // [B]
    const float* __restrict__ a,       // [B, L]
    const int* __restrict__ ebev,      // [B, L]
    float* __restrict__ fdist)         // [B, V+X]
{
  __shared__ float sred[256];
  const int b = blockIdx.x, t = threadIdx.x;
  const float* lg = logits + (size_t)b * Vn;
  float* fd = fdist + (size_t)b * (Vn + Xn);

  float lmax = -3.402823466e38f;
  for (int v = t; v < Vn; v += 256) lmax = fmaxf(lmax, lg[v]);
  sred[t] = lmax; __syncthreads();
  for (int s = 128; s > 0; s >>= 1) {
    if (t < s) sred[t] = fmaxf(sred[t], sred[t + s]);
    __syncthreads();
  }
  const float mx = sred[0]; __syncthreads();

  float lsum = 0.0f;
  for (int v = t; v < Vn; v += 256) lsum += __expf(lg[v] - mx);
  sred[t] = lsum; __syncthreads();
  for (int s = 128; s > 0; s >>= 1) {
    if (t < s) sred[t] += sred[t + s];
    __syncthreads();
  }
  const float sum = sred[0];

  const float pg = pgen[b];
  const float scale = pg / sum;
  for (int v = t; v < Vn + Xn; v += 256)
    fd[v] = (v < Vn) ? scale * __expf(lg[v] - mx) : 0.0f;

  __threadfence();
  __syncthreads();

  const float omp = 1.0f - pg;
  for (int l = t; l < Ln; l += 256)
    atomicAdd(&fd[ebev[b * Ln + l]], omp * a[b * Ln + l]);
}

// ---------------------------------------------------------------------------
extern "C" void kernel_launch(void* const* d_in, const int* in_sizes, int n_in,
                              void* d_out, int out_size, void* d_ws, size_t ws_size,
                              hipStream_t stream) {
  // inputs (setup_inputs order)
  const int*   y_t       = (const int*)  d_in[0];
  const float* h0        = (const float*)d_in[1];
  const float* c0        = (const float*)d_in[2];
  const float* enc_outs  = (const float*)d_in[3];
  const float* enc_feats = (const float*)d_in[4];
  const float* enc_mask  = (const float*)d_in[5];
  const float* c_t_1     = (const float*)d_in[6];
  /* extra_zeros d_in[7] unused (we write zeros directly) */
  const int*   ebev      = (const int*)  d_in[8];
  const float* coverage  = (const float*)d_in[9];
  const float* embed_W   = (const float*)d_in[10];
  const float* Wih       = (const float*)d_in[11];
  const float* Whh       = (const float*)d_in[12];
  const float* bih       = (const float*)d_in[13];
  const float* bhh       = (const float*)d_in[14];
  const float* xctx_W    = (const float*)d_in[15];
  const float* xctx_b    = (const float*)d_in[16];
  const float* Ws_W      = (const float*)d_in[17];
  const float* Ws_b      = (const float*)d_in[18];
  const float* Wc_W      = (const float*)d_in[19];
  const float* v_W       = (const float*)d_in[20];
  const float* pgen_W    = (const float*)d_in[21];
  const float* pgen_b    = (const float*)d_in[22];
  const float* proj_W    = (const float*)d_in[23];
  const float* proj_b    = (const float*)d_in[24];
  const float* out_W     = (const float*)d_in[25];
  const float* out_b     = (const float*)d_in[26];

  // outputs concatenated: final_dist, h, c, c_t, a, p_gen, coverage_next
  float* out   = (float*)d_out;
  float* fdist = out;
  float* h_o   = fdist + (size_t)Bn * (Vn + Xn);
  float* c_o   = h_o   + (size_t)Bn * Hn;
  float* ct_o  = c_o   + (size_t)Bn * Hn;
  float* a_o   = ct_o  + (size_t)Bn * 2 * Hn;
  float* pg_o  = a_o   + (size_t)Bn * Ln;
  float* cov_o = pg_o  + Bn;

  // workspace
  float* ws     = (float*)d_ws;
  float* xcat   = ws;  ws += (size_t)Bn * 640;     // [B, 2H+E]
  float* x      = ws;  ws += (size_t)Bn * En;      // [B, E]
  float* gates  = ws;  ws += (size_t)Bn * 1024;    // [B, 4H]
  float* sth    = ws;  ws += (size_t)Bn * 512;     // [B, 2H]
  float* decf   = ws;  ws += (size_t)Bn * 512;     // [B, 2H]
  float* pcat   = ws;  ws += (size_t)Bn * 768;     // [B, 3H]
  float* ohid   = ws;  ws += (size_t)Bn * Hn;      // [B, H]
  float* logits = ws;  ws += (size_t)Bn * Vn;      // [B, V]

  auto gemm = [&](const float* A, const float* Wt, const float* bias, float* C,
                  int M, int N, int K, int acc) {
    const int ntiles = N / 16;
    dim3 grid((ntiles + WPB - 1) / WPB, M / 16);
    size_t shmem = (size_t)16 * K * sizeof(float);   // A strip in LDS
    gemm_wmma_f32<<<grid, WPB * 32, shmem, stream>>>(A, Wt, bias, C, M, N, K, acc);
  };

  // 1. xcat = [c_t_1 | embed]
  build_xcat_kernel<<<Bn, 256, 0, stream>>>(y_t, c_t_1, embed_W, xcat);
  // 2. x = xcat @ xctx_W^T + xctx_b          [128,640]x[640,128]
  gemm(xcat, xctx_W, xctx_b, x, Bn, En, 640, 0);
  // 3. gates = x @ Wih^T + bih               [128,128]x[128,1024]
  gemm(x, Wih, bih, gates, Bn, 1024, En, 0);
  // 4. gates += h0 @ Whh^T + bhh             [128,256]x[256,1024]
  gemm(h0, Whh, bhh, gates, Bn, 1024, Hn, 1);
  // 5. LSTM cell -> h, c, s_t_hat
  lstm_kernel<<<Bn, 256, 0, stream>>>(gates, c0, h_o, c_o, sth);
  // 6. dec_feats = s_t_hat @ Ws_W^T + Ws_b   [128,512]x[512,512]
  gemm(sth, Ws_W, Ws_b, decf, Bn, 512, 512, 0);
  // 7. attention -> a, coverage_next, c_t
  attn_kernel<<<Bn, 256, 0, stream>>>(enc_feats, enc_outs, enc_mask, coverage,
                                      decf, Wc_W, v_W, a_o, ct_o, cov_o);
  // 8. p_gen
  pgen_kernel<<<Bn, 256, 0, stream>>>(ct_o, sth, x, pgen_W, pgen_b, pg_o);
  // 9. pcat = [h | c_t]
  build_pcat_kernel<<<Bn, 256, 0, stream>>>(h_o, ct_o, pcat);
  // 10. out_hid = pcat @ proj_W^T + proj_b   [128,768]x[768,256]
  gemm(pcat, proj_W, proj_b, ohid, Bn, Hn, 768, 0);
  // 11. logits = out_hid @ out_W^T + out_b   [128,256]x[256,50000]
  gemm(ohid, out_W, out_b, logits, Bn, Vn, Hn, 0);
  // 12. softmax * p_gen, extend, pointer scatter-add
  final_kernel<<<Bn, 256, 0, stream>>>(logits, pg_o, a_o, ebev, fdist);
}